// EstimatorQNN_65481071397771
// MI455X (gfx1250) — compile-verified
//
#include <hip/hip_runtime.h>

// ---------------------------------------------------------------------------
// Reference folds to: out[i] = a0 + a1*cos(theta) + a2*sin(theta),
// theta = in[i,0] + in[i,1]; a0,a1,a2 derived once from the 4 weights.
// Pure HBM-streaming kernel: 100.7 MB total -> ~4.3 us floor at 23.3 TB/s.
// Data path: async global->LDS B128 staging (ASYNCcnt, th:TH_LOAD_NT),
// B128 LDS reads, native v_sin/v_cos, non-temporal B128 stores.
// WMMA deliberately unused: after the algebraic fold there is no matmul
// structure left (the batch-independent 2x2 complex matvec reduces to 3
// scalar constants), so matrix ops would only waste VGPRs on a kernel whose
// ceiling is HBM bandwidth.
// ---------------------------------------------------------------------------

typedef float v4f __attribute__((ext_vector_type(4)));

#define TPB  256                 // 8 wave32s per block
#define EPT  8                   // elements per thread (64 B in, 32 B out)
#define TILE (TPB * EPT)         // 2048 elements = 16 KB of float2 input/block

namespace {
struct C2 { float re, im; };
__device__ inline C2 cmul(C2 a, C2 b) {
    return { a.re * b.re - a.im * b.im, a.re * b.im + a.im * b.re };
}
__device__ inline C2 cadd(C2 a, C2 b) { return { a.re + b.re, a.im + b.im }; }
__device__ inline float cnorm2(C2 a) { return a.re * a.re + a.im * a.im; }
} // namespace

// One thread: build U = RZ(w3)RX(w2) @ RZ(w1)RX(w0), fold to 3 coefficients.
// Native v_sin/v_cos (args are O(1) normal deviates / 2 — tiny, well inside
// the hardware range), avoids the scalarized Payne-Hanek blob.
__global__ void qnn_setup_kernel(const float* __restrict__ w,
                                 float* __restrict__ coef) {
    if (threadIdx.x != 0 || blockIdx.x != 0) return;

    C2 U[2][2] = { { {1.f, 0.f}, {0.f, 0.f} },
                   { {0.f, 0.f}, {1.f, 0.f} } };

    for (int l = 0; l < 2; ++l) {
        const float xh = 0.5f * w[2 * l];       // RX angle / 2
        const float zh = 0.5f * w[2 * l + 1];   // RZ angle / 2
        const float c  = __cosf(xh), s  = __sinf(xh);
        const float ce = __cosf(zh), se = __sinf(zh);
        // G = RZ(z) @ RX(x)
        C2 G[2][2];
        G[0][0] = {  c * ce, -c * se };   //  e^{-iz/2} * c
        G[0][1] = { -s * se, -s * ce };   // -i e^{-iz/2} * s
        G[1][0] = {  s * se, -s * ce };   // -i e^{+iz/2} * s
        G[1][1] = {  c * ce,  c * se };   //  e^{+iz/2} * c

        C2 R[2][2];
        for (int i = 0; i < 2; ++i)
            for (int j = 0; j < 2; ++j)
                R[i][j] = cadd(cmul(G[i][0], U[0][j]), cmul(G[i][1], U[1][j]));
        for (int i = 0; i < 2; ++i)
            for (int j = 0; j < 2; ++j) U[i][j] = R[i][j];
    }

    const float A = cnorm2(U[0][0]) - cnorm2(U[1][0]);
    const float B = cnorm2(U[0][1]) - cnorm2(U[1][1]);
    const float r0 = U[0][0].re * U[0][1].re + U[0][0].im * U[0][1].im; // Re(U00 U01*)
    const float r1 = U[1][0].re * U[1][1].re + U[1][0].im * U[1][1].im; // Re(U10 U11*)
    const float C = 2.f * (r0 - r1);

    coef[0] = 0.5f * (A + B);   // a0
    coef[1] = 0.5f * (A - B);   // a1 (multiplies cos theta)
    coef[2] = 0.5f * C;         // a2 (multiplies sin theta)
}

__global__ __launch_bounds__(TPB)
void qnn_main_kernel(const float* __restrict__ in,   // [n,2] flat
                     const float* __restrict__ coef, // a0,a1,a2
                     float* __restrict__ out,        // [n]
                     int n) {
    __shared__ float tile[TILE * 2] __attribute__((aligned(16)));  // 16 KB

    const int       tid      = threadIdx.x;
    const long long tileBase = (long long)blockIdx.x * TILE;

    const float a0 = coef[0], a1 = coef[1], a2 = coef[2];

    if (tileBase + TILE <= (long long)n) {
        // ---- async stage this block's 16 KB input tile into LDS ----------
        // LDS aperture keeps the wave-relative offset in addr[31:0], so
        // truncating the generic pointer yields the raw LDS byte address.
        // Input is single-use: stream it (th:TH_LOAD_NT), don't install in L2.
        const unsigned l0 = (unsigned)(size_t)(&tile[tid * 16]);
        const unsigned g0 = (unsigned)((tileBase + (long long)tid * EPT) * 8ll);
        asm volatile(
            "global_load_async_to_lds_b128 %0, %4, %8 th:TH_LOAD_NT\n\t"
            "global_load_async_to_lds_b128 %1, %5, %8 th:TH_LOAD_NT\n\t"
            "global_load_async_to_lds_b128 %2, %6, %8 th:TH_LOAD_NT\n\t"
            "global_load_async_to_lds_b128 %3, %7, %8 th:TH_LOAD_NT"
            :
            : "v"(l0), "v"(l0 + 16u), "v"(l0 + 32u), "v"(l0 + 48u),
              "v"(g0), "v"(g0 + 16u), "v"(g0 + 32u), "v"(g0 + 48u),
              "s"(in)
            : "memory");
        asm volatile("s_wait_asynccnt 0" ::: "memory");
        __syncthreads();

        // ---- consume: B128 LDS reads, 8 elements / thread ----------------
        const v4f q0 = *(const v4f*)(&tile[tid * 16]);       // x0 y0 x1 y1
        const v4f q1 = *(const v4f*)(&tile[tid * 16 + 4]);   // x2 y2 x3 y3
        const v4f q2 = *(const v4f*)(&tile[tid * 16 + 8]);   // x4 y4 x5 y5
        const v4f q3 = *(const v4f*)(&tile[tid * 16 + 12]);  // x6 y6 x7 y7

        const float t0 = q0.x + q0.y;
        const float t1 = q0.z + q0.w;
        const float t2 = q1.x + q1.y;
        const float t3 = q1.z + q1.w;
        const float t4 = q2.x + q2.y;
        const float t5 = q2.z + q2.w;
        const float t6 = q3.x + q3.y;
        const float t7 = q3.z + q3.w;

        v4f r0, r1;
        r0.x = fmaf(a1, __cosf(t0), fmaf(a2, __sinf(t0), a0));
        r0.y = fmaf(a1, __cosf(t1), fmaf(a2, __sinf(t1), a0));
        r0.z = fmaf(a1, __cosf(t2), fmaf(a2, __sinf(t2), a0));
        r0.w = fmaf(a1, __cosf(t3), fmaf(a2, __sinf(t3), a0));
        r1.x = fmaf(a1, __cosf(t4), fmaf(a2, __sinf(t4), a0));
        r1.y = fmaf(a1, __cosf(t5), fmaf(a2, __sinf(t5), a0));
        r1.z = fmaf(a1, __cosf(t6), fmaf(a2, __sinf(t6), a0));
        r1.w = fmaf(a1, __cosf(t7), fmaf(a2, __sinf(t7), a0));

        // Write-once output: stream past the caches (NT stores).
        v4f* o = (v4f*)(out + tileBase) + tid * 2;
        __builtin_nontemporal_store(r0, o);
        __builtin_nontemporal_store(r1, o + 1);
    } else {
        // Tail block (n not multiple of TILE): scalar path, no barrier.
        for (int k = 0; k < EPT; ++k) {
            const long long e = tileBase + (long long)tid * EPT + k;
            if (e < (long long)n) {
                const float t = in[2 * e] + in[2 * e + 1];
                out[e] = fmaf(a1, __cosf(t), fmaf(a2, __sinf(t), a0));
            }
        }
    }
}

extern "C" void kernel_launch(void* const* d_in, const int* in_sizes, int n_in,
                              void* d_out, int out_size, void* d_ws, size_t ws_size,
                              hipStream_t stream) {
    const float* in   = (const float*)d_in[0];  // [8388608, 2] fp32
    const float* w    = (const float*)d_in[1];  // [4] fp32
    float*       out  = (float*)d_out;          // [8388608] fp32
    float*       coef = (float*)d_ws;           // 3 floats of scratch

    const int n = in_sizes[0] / 2;

    qnn_setup_kernel<<<1, 1, 0, stream>>>(w, coef);

    const int blocks = (n + TILE - 1) / TILE;
    qnn_main_kernel<<<blocks, TPB, 0, stream>>>(in, coef, out, n);
}